// Sample_Concrete_12206297055675
// MI455X (gfx1250) — compile-verified
//
#include <hip/hip_runtime.h>
#include <cstdint>

// ---------------------------------------------------------------------------
// Sample_Concrete (training branch), G=3 softmax over innermost contiguous
// groups, K=1. Memory-bound streaming kernel: ~1.21 GB of traffic, ~52 us
// floor at 23.3 TB/s. No matmul -> WMMA inapplicable; the CDNA5 feature that
// matters here is the async global->LDS copy path + NT stores.
// ---------------------------------------------------------------------------

typedef float v4f __attribute__((ext_vector_type(4)));
typedef int   v4i __attribute__((vector_size(16)));   // matches builtin param

#define TINY_F   1.17549435e-38f   // np.finfo(np.float32).tiny
#define TAU_INV  2.0f              // 1 / 0.5

#define THREADS            256
#define FLOATS_PER_THREAD  12                       // 4 groups of 3
#define FLOATS_PER_BLOCK   (THREADS * FLOATS_PER_THREAD)   // 3072
#define CHUNKS_PER_BLOCK   (FLOATS_PER_BLOCK / 4)          // 768 x 16B

#define AS1 __attribute__((address_space(1)))
#define AS3 __attribute__((address_space(3)))

// 16-byte async copy: global -> LDS, tracked by ASYNCcnt.
__device__ __forceinline__ void async_cp16(v4f* lds_dst, const v4f* g_src) {
#if __has_builtin(__builtin_amdgcn_global_load_async_to_lds_b128)
  __builtin_amdgcn_global_load_async_to_lds_b128(
      (AS1 v4i*)g_src, (AS3 v4i*)lds_dst, /*imm offset*/0, /*cpol*/0);
#else
  unsigned loff = (unsigned)(uintptr_t)lds_dst;   // low 32 bits = LDS byte addr
  asm volatile("global_load_async_to_lds_b128 %0, %1, off"
               :: "v"(loff), "v"(g_src)
               : "memory");
#endif
}

__device__ __forceinline__ void wait_async_all() {
#if __has_builtin(__builtin_amdgcn_s_wait_asynccnt)
  __builtin_amdgcn_s_wait_asynccnt(0);
#else
  asm volatile("s_wait_asynccnt 0x0" ::: "memory");
#endif
}

__device__ __forceinline__ float gumbelize(float u) {
  u = __builtin_fminf(__builtin_fmaxf(u, TINY_F), 1.0f);
  return -__logf(-__logf(u));   // v_log_f32 x2 (native)
}

__device__ __forceinline__ void softmax3(float l0, float l1, float l2,
                                         float u0, float u1, float u2,
                                         float& y0, float& y1, float& y2) {
  float x0 = (l0 + gumbelize(u0)) * TAU_INV;
  float x1 = (l1 + gumbelize(u1)) * TAU_INV;
  float x2 = (l2 + gumbelize(u2)) * TAU_INV;
  float m  = __builtin_fmaxf(x0, __builtin_fmaxf(x1, x2));
  float e0 = __expf(x0 - m);
  float e1 = __expf(x1 - m);
  float e2 = __expf(x2 - m);
  float r  = __builtin_amdgcn_rcpf(e0 + e1 + e2);   // v_rcp_f32
  y0 = e0 * r;
  y1 = e1 * r;
  y2 = e2 * r;
}

__global__ __launch_bounds__(THREADS)
void sample_concrete_kernel(const float* __restrict__ logits,
                            const float* __restrict__ unif,
                            float* __restrict__ out,
                            size_t n) {
  __shared__ v4f sL[CHUNKS_PER_BLOCK];   // 12 KB logits tile
  __shared__ v4f sU[CHUNKS_PER_BLOCK];   // 12 KB uniform tile

  const int    t    = threadIdx.x;
  const size_t base = (size_t)blockIdx.x * FLOATS_PER_BLOCK;

  if (base + FLOATS_PER_BLOCK <= n) {
    // -------- fast path: async-staged, fully vectorized tile --------
    const v4f* gL = (const v4f*)(logits + base);
    const v4f* gU = (const v4f*)(unif   + base);

    #pragma unroll
    for (int k = 0; k < 3; ++k) {
      const int c = t + k * THREADS;          // coalesced 16B chunks
      async_cp16(&sL[c], gL + c);
      async_cp16(&sU[c], gU + c);
    }
    wait_async_all();      // this wave's async copies landed in LDS
    __syncthreads();       // make all waves' tiles visible

    float l[FLOATS_PER_THREAD], u[FLOATS_PER_THREAD], y[FLOATS_PER_THREAD];
    #pragma unroll
    for (int k = 0; k < 3; ++k) {
      v4f lv = sL[3 * t + k];                 // ds_load_b128
      v4f uv = sU[3 * t + k];
      #pragma unroll
      for (int j = 0; j < 4; ++j) {
        l[4 * k + j] = lv[j];
        u[4 * k + j] = uv[j];
      }
    }

    #pragma unroll
    for (int g = 0; g < 4; ++g)
      softmax3(l[3 * g], l[3 * g + 1], l[3 * g + 2],
               u[3 * g], u[3 * g + 1], u[3 * g + 2],
               y[3 * g], y[3 * g + 1], y[3 * g + 2]);

    v4f* ov = (v4f*)(out + base);
    #pragma unroll
    for (int k = 0; k < 3; ++k) {
      v4f r = { y[4 * k], y[4 * k + 1], y[4 * k + 2], y[4 * k + 3] };
      __builtin_nontemporal_store(r, ov + 3 * t + k);   // one-pass stream
    }
  } else {
    // -------- scalar tail (not taken with the reference shapes) --------
    for (size_t i = base + (size_t)t * 3; i + 2 < n;
         i += (size_t)THREADS * 3) {
      float y0, y1, y2;
      softmax3(logits[i], logits[i + 1], logits[i + 2],
               unif[i],   unif[i + 1],   unif[i + 2],
               y0, y1, y2);
      out[i]     = y0;
      out[i + 1] = y1;
      out[i + 2] = y2;
    }
  }
}

extern "C" void kernel_launch(void* const* d_in, const int* in_sizes, int n_in,
                              void* d_out, int out_size, void* d_ws, size_t ws_size,
                              hipStream_t stream) {
  const float* logits = (const float*)d_in[0];   // (B, F*G) fp32
  const float* unif   = (const float*)d_in[1];   // (B, F, 1, G) fp32, same flat layout
  float*       out    = (float*)d_out;           // (B, F*G) fp32

  const size_t n = (size_t)out_size;             // 100,663,296 for reference shapes
  const unsigned blocks =
      (unsigned)((n + FLOATS_PER_BLOCK - 1) / FLOATS_PER_BLOCK);   // 32768

  hipLaunchKernelGGL(sample_concrete_kernel, dim3(blocks), dim3(THREADS), 0,
                     stream, logits, unif, out, n);
}